// SAModule_66529043415497
// MI455X (gfx1250) — compile-verified
//
#include <hip/hip_runtime.h>
#include <hip/hip_bf16.h>

typedef float v2f __attribute__((ext_vector_type(2)));
typedef float v8f __attribute__((ext_vector_type(8)));

constexpr int Bc  = 16;     // clouds
constexpr int Np  = 4096;   // points per cloud
constexpr int Df  = 64;     // feature dim
constexpr int Hd  = 128;    // hidden
constexpr int Kn  = 32;     // max neighbors
constexpr int NSs = 1024;   // fps samples
constexpr float R2 = 0.04f; // radius^2

// ---------------------------------------------------------------------------
// Kernel 1: farthest point sampling, one block per cloud.
// One barrier per step: per-wave shuffle argmax, then every wave redundantly
// reduces the 32 wave candidates (butterfly leaves result in all lanes), so
// `last` is a uniform register. Candidate arrays double-buffered by parity.
// ---------------------------------------------------------------------------
__global__ __launch_bounds__(1024) void fps_kernel(const float* __restrict__ pos,
                                                   float* __restrict__ pos_s,
                                                   int* __restrict__ idx_out) {
    __shared__ float sPos[Np * 3];
    __shared__ float sRV[2][32];
    __shared__ int   sRI[2][32];

    const int b = blockIdx.x;
    const int tid = threadIdx.x;
    const int lane = tid & 31;
    const int wid  = tid >> 5;

    for (int i = tid; i < Np * 3; i += 1024) sPos[i] = pos[b * Np * 3 + i];

    float dl[4];
#pragma unroll
    for (int j = 0; j < 4; ++j) dl[j] = 1e10f;

    int last = 0;
    int buf  = 0;
    __syncthreads();

    for (int t = 0; t < NSs; ++t) {
        if (tid == 0) {
            idx_out[b * NSs + t] = last;
            pos_s[(b * NSs + t) * 3 + 0] = sPos[last * 3 + 0];
            pos_s[(b * NSs + t) * 3 + 1] = sPos[last * 3 + 1];
            pos_s[(b * NSs + t) * 3 + 2] = sPos[last * 3 + 2];
        }
        const float px = sPos[last * 3 + 0];
        const float py = sPos[last * 3 + 1];
        const float pz = sPos[last * 3 + 2];

        float bv = -1.0f;
        int   bi = 0;
#pragma unroll
        for (int j = 0; j < 4; ++j) {
            const int i = tid * 4 + j;
            const float dx = sPos[i * 3 + 0] - px;
            const float dy = sPos[i * 3 + 1] - py;
            const float dz = sPos[i * 3 + 2] - pz;
            const float d  = dx * dx + dy * dy + dz * dz;
            dl[j] = fminf(dl[j], d);
            if (dl[j] > bv) { bv = dl[j]; bi = i; }
        }
        // wave32 argmax (prefer smaller index on ties, like jnp.argmax)
#pragma unroll
        for (int m = 16; m >= 1; m >>= 1) {
            const float ov = __shfl_xor(bv, m, 32);
            const int   oi = __shfl_xor(bi, m, 32);
            if (ov > bv || (ov == bv && oi < bi)) { bv = ov; bi = oi; }
        }
        if (lane == 0) { sRV[buf][wid] = bv; sRI[buf][wid] = bi; }
        __syncthreads();
        // every wave reduces the 32 candidates -> uniform result in all lanes
        bv = sRV[buf][lane];
        bi = sRI[buf][lane];
#pragma unroll
        for (int m = 16; m >= 1; m >>= 1) {
            const float ov = __shfl_xor(bv, m, 32);
            const int   oi = __shfl_xor(bi, m, 32);
            if (ov > bv || (ov == bv && oi < bi)) { bv = ov; bi = oi; }
        }
        last = bi;
        buf ^= 1;
    }
}

// ---------------------------------------------------------------------------
// Kernel 2: radius top-K neighbor search. One thread per center,
// 128 centers per block (all in the same cloud), cloud pos cached in LDS.
// Invalid neighbor slots written as -1.
// ---------------------------------------------------------------------------
__global__ __launch_bounds__(128) void nbr_kernel(const float* __restrict__ pos,
                                                  const float* __restrict__ pos_s,
                                                  int* __restrict__ nbr) {
    __shared__ float sPos[Np * 3];
    __shared__ float bd[Kn * 128];
    __shared__ int   bix[Kn * 128];

    const int tid = threadIdx.x;
    const int c = blockIdx.x * 128 + tid;
    const int b = c / NSs;

    for (int i = tid; i < Np * 3; i += 128) sPos[i] = pos[b * Np * 3 + i];
    __syncthreads();

    const float cx = pos_s[c * 3 + 0];
    const float cy = pos_s[c * 3 + 1];
    const float cz = pos_s[c * 3 + 2];

    int cnt = 0;
    float worstv = -1.0f;
    int wslot = 0;

    for (int i = 0; i < Np; ++i) {
        const float dx = sPos[i * 3 + 0] - cx;
        const float dy = sPos[i * 3 + 1] - cy;
        const float dz = sPos[i * 3 + 2] - cz;
        const float d2 = dx * dx + dy * dy + dz * dz;
        if (d2 <= R2) {
            if (cnt < Kn) {
                bd[cnt * 128 + tid]  = d2;
                bix[cnt * 128 + tid] = i;
                ++cnt;
                if (cnt == Kn) {
                    worstv = -1.0f;
                    for (int k = 0; k < Kn; ++k) {
                        const float v = bd[k * 128 + tid];
                        if (v > worstv) { worstv = v; wslot = k; }
                    }
                }
            } else if (d2 < worstv) {
                bd[wslot * 128 + tid]  = d2;
                bix[wslot * 128 + tid] = i;
                worstv = -1.0f;
                for (int k = 0; k < Kn; ++k) {
                    const float v = bd[k * 128 + tid];
                    if (v > worstv) { worstv = v; wslot = k; }
                }
            }
        }
    }
    for (int k = 0; k < Kn; ++k)
        nbr[c * Kn + k] = (k < cnt) ? bix[k * 128 + tid] : -1;
}

// ---------------------------------------------------------------------------
// Kernel 3: fused gather + MLP1 (WMMA) + aggregate + MLP2 (WMMA).
// 256 threads (8 waves) per block; one block handles 16 centers.
// ---------------------------------------------------------------------------
constexpr int FEAT_S = 73;   // 68 used + pad (odd stride: conflict-free row reads)
constexpr int EMB_S  = 129;
constexpr int VEC_S  = 193;

__global__ __launch_bounds__(256) void fused_kernel(
    const float* __restrict__ x, const float* __restrict__ pos,
    const float* __restrict__ W1, const float* __restrict__ b1,
    const float* __restrict__ W2, const float* __restrict__ b2,
    const float* __restrict__ W3, const float* __restrict__ b3,
    const float* __restrict__ pos_s, const int* __restrict__ nbr,
    const int* __restrict__ idx_i,
    float* __restrict__ out_x, float* __restrict__ out_pos) {

    __shared__ float sW1[68 * 128];          // W1 padded with zero rows 65..67
    __shared__ float sB1[128];
    __shared__ float sB3[128];
    __shared__ float sW2[128 * 3];
    __shared__ float sFeat[32 * FEAT_S];     // [K=32][64 feat + dist + pad]
    __shared__ float sEmb[32 * EMB_S];       // masked relu(feat@W1+b1)
    __shared__ float sVec[16 * VEC_S];       // concat(x_d, aggr_x) per center
    __shared__ float sDiff[32 * 3];
    __shared__ float sPmsg[32 * 3];
    __shared__ float sValid[32];
    __shared__ float sPosC[3];
    __shared__ int   sNbr[32];

    const int tid  = threadIdx.x;
    const int lane = tid & 31;
    const int w    = tid >> 5;        // wave id 0..7  -> N-tile
    const int n    = lane & 15;       // lane within half
    const int half = lane >> 4;       // 0 or 1
    const int col  = w * 16 + n;      // output column this lane owns

    // ---- phase 0: stage weights ----
    for (int i = tid; i < 65 * 128; i += 256) sW1[i] = W1[i];
    for (int i = tid; i < 3 * 128;  i += 256) sW1[65 * 128 + i] = 0.f;
    if (tid < 128) { sB1[tid] = b1[tid]; sB3[tid] = b3[tid]; }
    for (int i = tid; i < 128 * 3; i += 256) sW2[i] = W2[i];
    if (tid < 96) {                    // feature pad columns 65..67 are always 0
        const int k = tid / 3, cc = 65 + tid % 3;
        sFeat[k * FEAT_S + cc] = 0.f;
    }
    __syncthreads();

    for (int gi = 0; gi < 16; ++gi) {
        const int c = blockIdx.x * 16 + gi;
        const int b = c / NSs;

        if (tid < 32) {
            const int nb = nbr[c * Kn + tid];
            sNbr[tid]   = (nb < 0) ? 0 : nb;
            sValid[tid] = (nb < 0) ? 0.f : 1.f;
        }
        if (tid < 3) sPosC[tid] = pos_s[c * 3 + tid];
        __syncthreads();

        if (tid < 32) {
            const int pj = sNbr[tid];
            const float dx = pos[(b * Np + pj) * 3 + 0] - sPosC[0];
            const float dy = pos[(b * Np + pj) * 3 + 1] - sPosC[1];
            const float dz = pos[(b * Np + pj) * 3 + 2] - sPosC[2];
            sDiff[tid * 3 + 0] = dx;
            sDiff[tid * 3 + 1] = dy;
            sDiff[tid * 3 + 2] = dz;
            const float d2 = dx * dx + dy * dy + dz * dz;
            sFeat[tid * FEAT_S + 64] = (d2 > 0.f) ? sqrtf(d2) : 0.f;
        }
        // gather neighbor features: 32*64 elems, coalesced over d
#pragma unroll
        for (int r = 0; r < 8; ++r) {
            const int e = tid + 256 * r;
            const int k = e >> 6, d = e & 63;
            sFeat[k * FEAT_S + d] = x[(b * Np + sNbr[k]) * Df + d];
        }
        __syncthreads();

        // ---- matmul1: emb[32,128] = relu(feat[32,68] @ W1[68,128] + b1), masked
        v8f acc0 = {};
        v8f acc1 = {};
        for (int kk = 0; kk < 17; ++kk) {
            const int kA = kk * 4 + 2 * half;
            v2f bf; bf.x = sW1[kA * 128 + col]; bf.y = sW1[(kA + 1) * 128 + col];
            v2f a0; a0.x = sFeat[n * FEAT_S + kA]; a0.y = sFeat[n * FEAT_S + kA + 1];
            v2f a1; a1.x = sFeat[(n + 16) * FEAT_S + kA]; a1.y = sFeat[(n + 16) * FEAT_S + kA + 1];
            acc0 = __builtin_amdgcn_wmma_f32_16x16x4_f32(false, a0, false, bf,
                                                         (short)0, acc0, false, false);
            acc1 = __builtin_amdgcn_wmma_f32_16x16x4_f32(false, a1, false, bf,
                                                         (short)0, acc1, false, false);
        }
        {
            const float bias = sB1[col];
#pragma unroll
            for (int v = 0; v < 8; ++v) {
                const int r0 = v + 8 * half;          // rows of M-tile 0
                const int r1 = 16 + v + 8 * half;     // rows of M-tile 1
                sEmb[r0 * EMB_S + col] = fmaxf(acc0[v] + bias, 0.f) * sValid[r0];
                sEmb[r1 * EMB_S + col] = fmaxf(acc1[v] + bias, 0.f) * sValid[r1];
            }
        }
        __syncthreads();

        // ---- aggregation (balanced across waves) ----
        if (tid < 96) {                                   // pw + pmsg: (k, j) pairs
            const int k = tid / 3, j = tid % 3;
            float p = b2[j];
            for (int cc = 0; cc < 128; ++cc)
                p += sEmb[k * EMB_S + cc] * sW2[cc * 3 + j];
            sPmsg[k * 3 + j] = sValid[k] * sDiff[k * 3 + j] * p;
        } else if (tid < 224) {                           // aggr_x (sum over K)
            const int cc = tid - 96;
            float ag = 0.f;
            for (int k = 0; k < 32; ++k) ag += sEmb[k * EMB_S + cc];
            sVec[gi * VEC_S + 64 + cc] = ag;
        } else {                                          // x_d gather (2 elems each)
            const int d  = tid - 224;
            const int ic = idx_i[c];
            sVec[gi * VEC_S + d]      = x[(b * Np + ic) * Df + d];
            sVec[gi * VEC_S + d + 32] = x[(b * Np + ic) * Df + d + 32];
        }
        __syncthreads();
        if (tid < 3) {
            float ap = 0.f, cv = 0.f;
            for (int k = 0; k < 32; ++k) { ap += sPmsg[k * 3 + tid]; cv += sValid[k]; }
            out_pos[c * 3 + tid] = sPosC[tid] + ap / fmaxf(cv, 1.f);
        }
        __syncthreads();
    }

    // ---- matmul2: out_x[16,128] = relu(vec[16,192] @ W3[192,128] + b3)
    v8f acc = {};
    for (int kk = 0; kk < 48; ++kk) {
        const int kA = kk * 4 + 2 * half;
        v2f a;  a.x  = sVec[n * VEC_S + kA];  a.y  = sVec[n * VEC_S + kA + 1];
        v2f bf; bf.x = W3[kA * 128 + col];    bf.y = W3[(kA + 1) * 128 + col];
        acc = __builtin_amdgcn_wmma_f32_16x16x4_f32(false, a, false, bf,
                                                    (short)0, acc, false, false);
    }
    {
        const float bias = sB3[col];
#pragma unroll
        for (int v = 0; v < 8; ++v) {
            const int r = v + 8 * half;               // center gi = r
            out_x[(blockIdx.x * 16 + r) * Hd + col] = fmaxf(acc[v] + bias, 0.f);
        }
    }
}

// ---------------------------------------------------------------------------
extern "C" void kernel_launch(void* const* d_in, const int* in_sizes, int n_in,
                              void* d_out, int out_size, void* d_ws, size_t ws_size,
                              hipStream_t stream) {
    const float* x   = (const float*)d_in[0];
    const float* pos = (const float*)d_in[1];
    const float* W1  = (const float*)d_in[2];
    const float* b1  = (const float*)d_in[3];
    const float* W2  = (const float*)d_in[4];
    const float* b2  = (const float*)d_in[5];
    const float* W3  = (const float*)d_in[6];
    const float* b3  = (const float*)d_in[7];

    float* out    = (float*)d_out;
    float* out_x  = out;                                   // [B,NS,H]
    float* out_p  = out + (size_t)Bc * NSs * Hd;           // [B,NS,3]
    int*   idx_i  = (int*)(out + (size_t)Bc * NSs * Hd + (size_t)Bc * NSs * 3); // [B,NS]

    float* pos_s = (float*)d_ws;                                    // B*NS*3 floats
    int*   nbr   = (int*)((char*)d_ws + (size_t)Bc * NSs * 3 * sizeof(float)); // B*NS*K

    fps_kernel<<<Bc, 1024, 0, stream>>>(pos, pos_s, idx_i);
    nbr_kernel<<<(Bc * NSs) / 128, 128, 0, stream>>>(pos, pos_s, nbr);
    fused_kernel<<<(Bc * NSs) / 16, 256, 0, stream>>>(x, pos, W1, b1, W2, b2, W3, b3,
                                                      pos_s, nbr, idx_i, out_x, out_p);
}